// MarginalCalibrationError_1529008357638
// MI455X (gfx1250) — compile-verified
//
#include <hip/hip_runtime.h>
#include <math.h>

#define NBINS 15
#define NCLS  100
#define NSEG  (NBINS * NCLS)   // 1500 segments
#define F4_PER_ROW (NCLS / 4)  // 25

typedef __attribute__((ext_vector_type(2))) float v2f;
typedef __attribute__((ext_vector_type(8))) float v8f;

// ---------------------------------------------------------------------------
// Kernel 1: zero the global accumulator (3 * 1500 floats in d_ws)
// ---------------------------------------------------------------------------
__global__ void mce_zero_kernel(float* __restrict__ g) {
    int i = blockIdx.x * blockDim.x + threadIdx.x;
    if (i < 3 * NSEG) g[i] = 0.0f;
}

// ---------------------------------------------------------------------------
// Kernel 2: bandwidth-bound histogram.
// Coalesced float4 loads (rows are 25 aligned float4s), LDS per-block
// histogram via ds_add_f32 atomics, merged to global via global_atomic_add_f32.
// ---------------------------------------------------------------------------
__global__ void mce_hist_kernel(const float* __restrict__ probas,
                                const int*  __restrict__ labels,
                                float* __restrict__ gacc,
                                int nrows) {
    __shared__ float s_cnt[NSEG];
    __shared__ float s_conf[NSEG];
    __shared__ float s_acc[NSEG];
    for (int i = threadIdx.x; i < NSEG; i += blockDim.x) {
        s_cnt[i] = 0.0f; s_conf[i] = 0.0f; s_acc[i] = 0.0f;
    }
    __syncthreads();

    const float step = 1.0f / 15.0f;          // linspace delta
    const int total4 = nrows * F4_PER_ROW;    // 12.5M, fits in int
    const int stride = gridDim.x * blockDim.x;

    for (int q = blockIdx.x * blockDim.x + threadIdx.x; q < total4; q += stride) {
        const float4 pv = reinterpret_cast<const float4*>(probas)[q];
        const int row = q / F4_PER_ROW;           // const-div -> mul
        const int col = (q - row * F4_PER_ROW) * 4;
        const int lab = labels[row];
        const float pe[4] = {pv.x, pv.y, pv.z, pv.w};
#pragma unroll
        for (int i = 0; i < 4; ++i) {
            const float p = pe[i];
            int b = (int)floorf(p * 15.0f);
            b = b < 0 ? 0 : (b > NBINS - 1 ? NBINS - 1 : b);
            const float lo = (float)b * step;
            const float hi = (float)(b + 1) * step;
            if (p > lo && p < hi) {               // strict edges, matches ref
                const int c = col + i;
                const int s = c * NBINS + b;
                atomicAdd(&s_cnt[s], 1.0f);
                atomicAdd(&s_conf[s], p);
                if (lab == c) atomicAdd(&s_acc[s], 1.0f);
            }
        }
    }
    __syncthreads();

    for (int i = threadIdx.x; i < NSEG; i += blockDim.x) {
        const float c0 = s_cnt[i], c1 = s_conf[i], c2 = s_acc[i];
        if (c0 != 0.0f) atomicAdd(&gacc[i],            c0);
        if (c1 != 0.0f) atomicAdd(&gacc[NSEG + i],     c1);
        if (c2 != 0.0f) atomicAdd(&gacc[2 * NSEG + i], c2);
    }
}

// ---------------------------------------------------------------------------
// Kernel 3: finalize (one wave32). Each lane computes sq_ce for classes
// lane, lane+32, lane+64, lane+96; the 32 partials are reduced wave-wide
// with V_WMMA_F32_16X16X4_F32 (B = ones): D[m][*] = p_m + p_{m+16}; summing
// the 8 D VGPRs per lane + one cross-half shuffle gives the full sum.
// ---------------------------------------------------------------------------
__global__ void mce_final_kernel(const float* __restrict__ gacc,
                                 float* __restrict__ out) {
    const int lane = threadIdx.x;  // 32 threads, EXEC all ones

    float partial = 0.0f;
    for (int c = lane; c < NCLS; c += 32) {
        float ns[NBINS], cs[NBINS], as[NBINS];
        float tot = 0.0f;
#pragma unroll
        for (int b = 0; b < NBINS; ++b) {
            ns[b] = gacc[c * NBINS + b];
            cs[b] = gacc[NSEG + c * NBINS + b];
            as[b] = gacc[2 * NSEG + c * NBINS + b];
            tot += ns[b];
        }
        const float totd = fmaxf(tot, 1.0f);
        float sq = 0.0f;
#pragma unroll
        for (int b = 0; b < NBINS; ++b) {
            const float n     = ns[b];
            const float denom = fmaxf(n, 1.0f);
            const float confs = cs[b] / denom;
            const float accs  = as[b] / denom;
            const float pb    = n / totd;
            const float d     = fabsf(confs - accs);
            const float term  = pb * d * d;     // p == 2
            if (n > 0.0f) sq += term;           // mask empty bins
        }
        partial += sq;
    }

    // Wave-wide sum via WMMA f32 16x16x4 with ones B-matrix.
    v2f a;  a[0] = partial; a[1] = 0.0f;
    v2f bm; bm[0] = 1.0f;   bm[1] = 1.0f;
    v8f cm = {};
    v8f dm = __builtin_amdgcn_wmma_f32_16x16x4_f32(
        /*neg_a=*/false, a, /*neg_b=*/false, bm,
        /*c_mod=*/(short)0, cm, /*reuse_a=*/false, /*reuse_b=*/false);

    float s8 = dm[0] + dm[1] + dm[2] + dm[3] + dm[4] + dm[5] + dm[6] + dm[7];
    // lanes 0-15 hold sum of M=0..7,16..23 ; lanes 16-31 hold M=8..15,24..31
    float total = __shfl(s8, 0, 32) + __shfl(s8, 16, 32);

    if (lane == 0) out[0] = sqrtf(total / (float)NCLS);
}

// ---------------------------------------------------------------------------
extern "C" void kernel_launch(void* const* d_in, const int* in_sizes, int n_in,
                              void* d_out, int out_size, void* d_ws, size_t ws_size,
                              hipStream_t stream) {
    const float* probas = (const float*)d_in[0];
    const int*   labels = (const int*)d_in[1];
    float* out  = (float*)d_out;
    float* gacc = (float*)d_ws;              // 3*1500 floats = 18 KB
    const int nrows = in_sizes[1];           // N = 500000; C fixed at 100

    mce_zero_kernel<<<(3 * NSEG + 255) / 256, 256, 0, stream>>>(gacc);
    mce_hist_kernel<<<1024, 256, 0, stream>>>(probas, labels, gacc, nrows);
    mce_final_kernel<<<1, 32, 0, stream>>>(gacc, out);
}